// StatsPool_62689342652990
// MI455X (gfx1250) — compile-verified
//
#include <hip/hip_runtime.h>
#include <hip/hip_bf16.h>

typedef float __attribute__((ext_vector_type(2))) v2f;
typedef float __attribute__((ext_vector_type(8))) v8f;

#define C_DIM 256
#define W_DIM 65536
#define L_DIM 512
#define BINSZ 128   /* W_DIM / L_DIM */

// One workgroup per output bin (128 consecutive w positions of batch 0).
// 8 waves/workgroup; wave i computes o[w] for its 16-w subtile via a chain of
// 64 V_WMMA_F32_16X16X4_F32 ops over the 256-channel contraction, then the
// block reduces sum(o) and sum(o^2) and emits latent[bin] = mean + std*eps.
__global__ __launch_bounds__(256) void StatsPool_wmma_f32(
    const float* __restrict__ x,       // [B,C,1,W] -- only batch 0 is read
    const float* __restrict__ conv_w,  // [1,C]
    const float* __restrict__ conv_b,  // [1]
    const float* __restrict__ eps,     // [L]
    float* __restrict__ out)           // [1,L]
{
    __shared__ float s_w[C_DIM];
    __shared__ float s_s1[8];
    __shared__ float s_s2[8];

    const int tid  = threadIdx.x;
    const int lane = tid & 31;
    const int wave = tid >> 5;
    const int bin  = blockIdx.x;

    // Stage the 256 channel weights in LDS (blockDim.x == C_DIM).
    s_w[tid] = conv_w[tid];
    __syncthreads();

    // WMMA operand lane mapping (wave32, 16x16x4 f32):
    //   A (16x4, rows all equal to the weight slice): VGPR0 = K0|K2, VGPR1 = K1|K3
    //   B (4x16 data tile):                           VGPR0 = K0|K2, VGPR1 = K1|K3
    const int n    = lane & 15;              // column (w offset within subtile)
    const int krow = (lane >> 4) << 1;       // 0 for lanes 0-15, 2 for lanes 16-31
    const int wb   = bin * BINSZ + wave * 16;
    const float* xcol = x + (size_t)wb + n;  // batch 0: x[c*W + w]

    v8f acc = {};
    #pragma unroll 8
    for (int cb = 0; cb < C_DIM; cb += 4) {
        const float* p = xcol + (size_t)(cb + krow) * W_DIM;
        v2f b;
        b.x = p[0];           // channel cb+krow
        b.y = p[W_DIM];       // channel cb+krow+1
        v2f a;
        a.x = s_w[cb + krow];
        a.y = s_w[cb + krow + 1];
        acc = __builtin_amdgcn_wmma_f32_16x16x4_f32(
            /*neg_a=*/false, a, /*neg_b=*/false, b,
            /*c_mod=*/(short)0, acc, /*reuse_a=*/false, /*reuse_b=*/false);
    }

    // Every D row equals the output row (A rows identical); VGPR0 carries
    // o[wb+n] on lanes 0-15 (M=0) and duplicates on lanes 16-31 (M=8).
    float o  = acc[0] + conv_b[0];
    float s1 = o;
    float s2 = o * o;
    // Full 32-lane reduction counts each of the 16 values twice -> halve.
    #pragma unroll
    for (int off = 16; off >= 1; off >>= 1) {
        s1 += __shfl_xor(s1, off, 32);
        s2 += __shfl_xor(s2, off, 32);
    }
    if (lane == 0) {
        s_s1[wave] = s1 * 0.5f;
        s_s2[wave] = s2 * 0.5f;
    }
    __syncthreads();

    if (tid == 0) {
        float S1 = 0.0f, S2 = 0.0f;
        #pragma unroll
        for (int i = 0; i < 8; ++i) { S1 += s_s1[i]; S2 += s_s2[i]; }
        const float inv = 1.0f / (float)BINSZ;
        float mean = S1 * inv;
        float msq  = S2 * inv;
        float var  = fmaxf(msq - mean * mean, 0.0f);
        out[bin] = mean + sqrtf(var) * eps[bin];
    }
}

extern "C" void kernel_launch(void* const* d_in, const int* in_sizes, int n_in,
                              void* d_out, int out_size, void* d_ws, size_t ws_size,
                              hipStream_t stream) {
    const float* x      = (const float*)d_in[0];
    const float* conv_w = (const float*)d_in[1];
    const float* conv_b = (const float*)d_in[2];
    const float* eps    = (const float*)d_in[3];
    float* out = (float*)d_out;

    StatsPool_wmma_f32<<<dim3(L_DIM), dim3(256), 0, stream>>>(x, conv_w, conv_b, eps, out);
}